// VMD_77464030151050
// MI455X (gfx1250) — compile-verified
//
#include <hip/hip_runtime.h>
#include <hip/hip_bf16.h>

typedef __attribute__((ext_vector_type(2))) float v2f;
typedef __attribute__((ext_vector_type(8))) float v8f;

#define NTOT (32u * 512u * 128u)   // elements per output array

__device__ inline v8f wmma4(v2f a, v2f b, v8f c) {
    // V_WMMA_F32_16X16X4_F32 : D(16x16) = A(16x4) * B(4x16) + C
    return __builtin_amdgcn_wmma_f32_16x16x4_f32(false, a, false, b, (short)0, c,
                                                 false, false);
}

// 512-point complex FFT for one wave32, four-step: 512 = 16 x 32.
//   Stage1: Y[k1][n2]  = sum_n1 F16[k1][n1] * in[32*n1 + n2]      (WMMA)
//   Stage2: Y'[k1][n2] = Y * Tw[k1][n2]                           (VALU)
//   Stage3: Z[k1][k2]  = sum_n2 Y'[k1][n2] * F32[n2][k2]          (WMMA)
//   out[(k1 + 16*k2) ^ xorOut] = Z * scale
// Tables hold the forward kernel e^{-i t} as (cos, -sin); csgn = +1 forward,
// csgn = -1 conjugates everything (inverse); caller applies 1/512 via scale.
__device__ void fft512(const float* inr, const float* ini,
                       float* outr, float* outi,
                       unsigned xorOut, float scale, float csgn,
                       const float* F16r, const float* F16i,
                       const float* Twr,  const float* Twi,
                       const float* F32r, const float* F32i,
                       float* yre, float* yim, int lane)
{
    const int nlo = lane & 15;
    const int hi  = lane >> 4;

    // ---- Stage 1 + twiddle: two column tiles (n2 in [c, c+16)) ----
    for (int c = 0; c < 32; c += 16) {
        v8f accr = {}; v8f acci = {};
#pragma unroll
        for (int j = 0; j < 4; ++j) {
            v2f Ar, Ai, nAi, Br, Bi;
#pragma unroll
            for (int v = 0; v < 2; ++v) {
                int kl = 4 * j + v + 2 * hi;          // local K (n1) 0..15
                int ia = nlo * 16 + kl;               // F16[k1 = nlo][n1 = kl]
                Ar[v] = F16r[ia];
                Ai[v] = csgn * F16i[ia];
                int ib = 32 * kl + c + nlo;           // in[32*n1 + n2]
                Br[v] = inr[ib];
                Bi[v] = ini[ib];
            }
            nAi[0] = -Ai[0]; nAi[1] = -Ai[1];
            accr = wmma4(Ar,  Br, accr);
            accr = wmma4(nAi, Bi, accr);              // -Ai*Bi
            acci = wmma4(Ar,  Bi, acci);
            acci = wmma4(Ai,  Br, acci);
        }
#pragma unroll
        for (int r = 0; r < 8; ++r) {
            int M = r + 8 * hi;                       // k1
            int N = c + nlo;                          // n2
            float tr = Twr[M * 32 + N];
            float ti = csgn * Twi[M * 32 + N];
            float vr = accr[r], vi = acci[r];
            yre[M * 32 + N] = vr * tr - vi * ti;
            yim[M * 32 + N] = vr * ti + vi * tr;
        }
    }
    __builtin_amdgcn_wave_barrier();

    // ---- Stage 3: two k2 tiles ----
    for (int c2 = 0; c2 < 32; c2 += 16) {
        v8f accr = {}; v8f acci = {};
#pragma unroll
        for (int j = 0; j < 8; ++j) {
            v2f Ar, Ai, nAi, Br, Bi;
#pragma unroll
            for (int v = 0; v < 2; ++v) {
                int kl = 4 * j + v + 2 * hi;          // n2 chunk 0..31
                int ia = nlo * 32 + kl;               // Y'[k1 = nlo][n2 = kl]
                Ar[v] = yre[ia];
                Ai[v] = yim[ia];
                int ib = kl * 32 + c2 + nlo;          // F32[n2][k2]
                Br[v] = F32r[ib];
                Bi[v] = csgn * F32i[ib];
            }
            nAi[0] = -Ai[0]; nAi[1] = -Ai[1];
            accr = wmma4(Ar,  Br, accr);
            accr = wmma4(nAi, Bi, accr);
            acci = wmma4(Ar,  Bi, acci);
            acci = wmma4(Ai,  Br, acci);
        }
#pragma unroll
        for (int r = 0; r < 8; ++r) {
            int k1 = r + 8 * hi;
            int k2 = c2 + nlo;
            int kk = (k1 + 16 * k2) ^ (int)xorOut;
            outr[kk] = accr[r] * scale;
            outi[kk] = acci[r] * scale;
        }
    }
    __builtin_amdgcn_wave_barrier();
}

// One wave32 per (b, d) series. All state in LDS.
__global__ __launch_bounds__(32)
void vmd_main(const float* __restrict__ x, float* __restrict__ out,
              float* __restrict__ omega_ws)
{
    __shared__ float F16r[256],  F16i[256];
    __shared__ float Twr[512],   Twi[512];
    __shared__ float F32r[1024], F32i[1024];
    __shared__ float xh_r[512],  xh_i[512];              // fftshift(fft(x))
    __shared__ float U0r[512], U0i[512], U1r[512], U1i[512]; // shifted spectra
    __shared__ float lam[512], xs[512];
    __shared__ float ar_[512], ai_[512], br_[512], bi_[512]; // fft in/out
    __shared__ float yre[512], yim[512];                 // fft mid scratch

    const int lane = threadIdx.x;
    const int b = blockIdx.x >> 7;
    const int d = blockIdx.x & 127;
    const float* xp = x + (size_t)b * 512 * 128 + d;

    const float PI2 = 6.28318530717958647692f;
    // Forward DFT tables: e^{-i 2pi p / N} = (cos, -sin)
    for (int idx = lane; idx < 256; idx += 32) {
        int k1 = idx >> 4, n1 = idx & 15;
        float s, c; __sincosf(PI2 * (float)((k1 * n1) & 15) * (1.0f / 16.0f), &s, &c);
        F16r[idx] = c; F16i[idx] = -s;
    }
    for (int idx = lane; idx < 512; idx += 32) {
        int k1 = idx >> 5, n2 = idx & 31;
        float s, c; __sincosf(PI2 * (float)(k1 * n2) * (1.0f / 512.0f), &s, &c);
        Twr[idx] = c; Twi[idx] = -s;
    }
    for (int idx = lane; idx < 1024; idx += 32) {
        int n2 = idx >> 5, k2 = idx & 31;
        float s, c; __sincosf(PI2 * (float)((n2 * k2) & 31) * (1.0f / 32.0f), &s, &c);
        F32r[idx] = c; F32i[idx] = -s;
    }

    // Load series, init state
#pragma unroll
    for (int t = 0; t < 16; ++t) {
        int n = lane + 32 * t;
        float v = xp[(size_t)n * 128];
        xs[n] = v; ar_[n] = v; ai_[n] = 0.0f;
        U0r[n] = 0.0f; U0i[n] = 0.0f; U1r[n] = 0.0f; U1i[n] = 0.0f;
        lam[n] = 0.0f;
    }
    __builtin_amdgcn_wave_barrier();

    // x_hat (shifted order via xorOut = 256; fftshift == ifftshift for N=512)
    fft512(ar_, ai_, xh_r, xh_i, 256u, 1.0f, 1.0f,
           F16r, F16i, Twr, Twi, F32r, F32i, yre, yim, lane);

    float om0 = 0.0f, om1 = 0.0f;

    for (int it = 0; it < 50; ++it) {
        // ---- mode updates (k=0 uses old U1; k=1 uses fresh U0) ----
        for (int k = 0; k < 2; ++k) {
            float omk = (k == 0) ? om0 : om1;
            const float* Uor = (k == 0) ? U1r : U0r;
            const float* Uoi = (k == 0) ? U1i : U0i;
            float* Ukr = (k == 0) ? U0r : U1r;
            float* Uki = (k == 0) ? U0i : U1i;
#pragma unroll
            for (int t = 0; t < 16; ++t) {
                int i = lane + 32 * t;
                float nr = xh_r[i] - Uor[i] + 0.5f * lam[i];  // lam added bin-wise
                float ni = xh_i[i] - Uoi[i];
                float f  = (float)(i - 256) * (1.0f / 512.0f);
                float df = f - omk;
                float inv = 1.0f / (1.0f + 1600.0f * df * df); // 2*ALPHA = 1600
                ar_[i] = nr * inv; ai_[i] = ni * inv;
            }
            __builtin_amdgcn_wave_barrier();
            // Hermitian symmetrization == fft(real(ifft(.)))
#pragma unroll
            for (int t = 0; t < 16; ++t) {
                int i = lane + 32 * t;
                int m = (512 - i) & 511;                      // mirror of shifted f
                Ukr[i] = 0.5f * (ar_[i] + ar_[m]);
                Uki[i] = 0.5f * (ai_[i] - ai_[m]);
            }
            __builtin_amdgcn_wave_barrier();
        }

        // ---- omega updates (half-spectrum power, f >= 0 <=> i >= 256) ----
        {
            float n0 = 0.0f, d0 = 0.0f, n1 = 0.0f, d1 = 0.0f;
#pragma unroll
            for (int t = 8; t < 16; ++t) {
                int i = lane + 32 * t;
                float f = (float)(i - 256) * (1.0f / 512.0f);
                float p0 = U0r[i] * U0r[i] + U0i[i] * U0i[i];
                float p1 = U1r[i] * U1r[i] + U1i[i] * U1i[i];
                n0 += f * p0; d0 += p0;
                n1 += f * p1; d1 += p1;
            }
            for (int off = 16; off >= 1; off >>= 1) {
                n0 += __shfl_xor(n0, off, 32);
                d0 += __shfl_xor(d0, off, 32);
                n1 += __shfl_xor(n1, off, 32);
                d1 += __shfl_xor(d1, off, 32);
            }
            om0 = n0 / (d0 + 1e-7f);
            om1 = n1 / (d1 + 1e-7f);
        }

        // ---- lam update: s = real(ifft(ifftshift(U0 + U1))) ----
#pragma unroll
        for (int t = 0; t < 16; ++t) {
            int kk = lane + 32 * t;
            int i = kk ^ 256;                                 // ifftshift
            ar_[kk] = U0r[i] + U1r[i];
            ai_[kk] = U0i[i] + U1i[i];
        }
        __builtin_amdgcn_wave_barrier();
        fft512(ar_, ai_, br_, bi_, 0u, 1.0f / 512.0f, -1.0f,
               F16r, F16i, Twr, Twi, F32r, F32i, yre, yim, lane);
#pragma unroll
        for (int t = 0; t < 16; ++t) {
            int n = lane + 32 * t;
            lam[n] += 0.001f * (xs[n] - br_[n]);              // TAU = 0.001
        }
        __builtin_amdgcn_wave_barrier();
    }

    // ---- final time-domain modes -> out slots 1 (u0) and 2 (u1) ----
    const size_t base = (size_t)b * 512 * 128 + d;
    for (int k = 0; k < 2; ++k) {
        const float* Ukr = (k == 0) ? U0r : U1r;
        const float* Uki = (k == 0) ? U0i : U1i;
#pragma unroll
        for (int t = 0; t < 16; ++t) {
            int kk = lane + 32 * t;
            int i = kk ^ 256;
            ar_[kk] = Ukr[i]; ai_[kk] = Uki[i];
        }
        __builtin_amdgcn_wave_barrier();
        fft512(ar_, ai_, br_, bi_, 0u, 1.0f / 512.0f, -1.0f,
               F16r, F16i, Twr, Twi, F32r, F32i, yre, yim, lane);
        float* op = out + (size_t)NTOT * (size_t)(1 + k) + base;
#pragma unroll
        for (int t = 0; t < 16; ++t) {
            int n = lane + 32 * t;
            op[(size_t)n * 128] = br_[n];
        }
    }

    if (lane == 0) {
        omega_ws[(b * 2 + 0) * 128 + d] = om0;
        omega_ws[(b * 2 + 1) * 128 + d] = om1;
    }
}

// Per-batch stable argsort over K=2 of mean_d(omega): swap iff m0 > m1.
__global__ void vmd_order(const float* __restrict__ omega_ws, int* __restrict__ flags)
{
    int b = threadIdx.x;
    if (b < 32) {
        float m0 = 0.0f, m1 = 0.0f;
        for (int d = 0; d < 128; ++d) {
            m0 += omega_ws[(b * 2 + 0) * 128 + d];
            m1 += omega_ws[(b * 2 + 1) * 128 + d];
        }
        flags[b] = (m0 <= m1) ? 0 : 1;
    }
}

// Zero trend slot, conditionally swap period/residual slots per batch.
__global__ void vmd_finalize(float* __restrict__ out, const int* __restrict__ flags)
{
    unsigned idx = blockIdx.x * 256u + threadIdx.x;
    if (idx < NTOT) {
        out[idx] = 0.0f;                                      // x_trend
        int b = (int)(idx / (512u * 128u));
        if (flags[b]) {
            float v1 = out[NTOT + idx];
            float v2 = out[2u * NTOT + idx];
            out[NTOT + idx]      = v2;
            out[2u * NTOT + idx] = v1;
        }
    }
}

extern "C" void kernel_launch(void* const* d_in, const int* in_sizes, int n_in,
                              void* d_out, int out_size, void* d_ws, size_t ws_size,
                              hipStream_t stream)
{
    const float* x = (const float*)d_in[0];
    float* out = (float*)d_out;
    float* omega = (float*)d_ws;                              // 32*2*128 floats = 32 KB
    int*   flags = (int*)((char*)d_ws + 32 * 2 * 128 * sizeof(float));

    vmd_main<<<4096, 32, 0, stream>>>(x, out, omega);
    vmd_order<<<1, 32, 0, stream>>>(omega, flags);
    vmd_finalize<<<(NTOT + 255u) / 256u, 256, 0, stream>>>(out, flags);
}